// LPrompt_41747082117551
// MI455X (gfx1250) — compile-verified
//
#include <hip/hip_runtime.h>
#include <math.h>

typedef __attribute__((ext_vector_type(2))) float v2f;
typedef __attribute__((ext_vector_type(8))) float v8f;

// Problem constants (match reference)
#define BB 32
#define NN 2048
#define DD 1024
#define SPLITS 16           // N-splits for mean reduction
#define NCPT 10
#define TOPK 3
#define HEADS 16
#define GRP 3
#define HD 64

// d_out flat offsets (return-tuple order)
#define OFF_BP   0          // batched_prompt (2,32,16,3,64) = 196608
#define OFF_RS2  196608
#define OFF_RS   196609
#define OFF_SIM  196610     // (32,10)
#define OFF_DESC 196930     // (32,64)
#define OFF_PKN  198978     // (10,1024)
#define OFF_XN   209218     // (32,1024)

// ws float offsets
#define WS_PART  0                          // 32*16*1024 = 524288
#define WS_XN    524288                     // 32768
#define WS_PKN   557056                     // 10240
#define WS_PK2   567296                     // 1024
#define WS_BPR   568320                     // 2048
#define WS_RS2P  570368                     // 32
#define WS_RSP   570400                     // 32

__device__ __forceinline__ float blockReduceSum256(float v, float* sdata) {
    int t = threadIdx.x;
    sdata[t] = v;
    __syncthreads();
    for (int s = 128; s > 0; s >>= 1) {
        if (t < s) sdata[t] += sdata[t + s];
        __syncthreads();
    }
    float r = sdata[0];
    __syncthreads();
    return r;
}

// ---------------- Kernel A: split-N mean partials (bandwidth bound) -------------
__global__ __launch_bounds__(256) void mean_partial_kernel(const float* __restrict__ x,
                                                           float* __restrict__ partial) {
    int blk = blockIdx.x;              // b * SPLITS + s
    int b = blk / SPLITS, s = blk % SPLITS;
    int t = threadIdx.x;
    const int rows = NN / SPLITS;      // 128
    const float* base = x + ((size_t)b * NN + (size_t)s * rows) * DD + 4 * t;
    float4 acc = make_float4(0.f, 0.f, 0.f, 0.f);
#pragma unroll 4
    for (int n = 0; n < rows; n++) {
        float4 v = *(const float4*)(base + (size_t)n * DD);
        acc.x += v.x; acc.y += v.y; acc.z += v.z; acc.w += v.w;
    }
    *(float4*)(partial + (size_t)blk * DD + 4 * t) = acc;
}

// ---------------- Kernel B: combine partials, mean, L2-normalize ---------------
__global__ __launch_bounds__(256) void finalize_norm_kernel(const float* __restrict__ partial,
                                                            float* __restrict__ xnorm,
                                                            float* __restrict__ out) {
    __shared__ float red[256];
    int b = blockIdx.x, t = threadIdx.x;
    float4 m = make_float4(0.f, 0.f, 0.f, 0.f);
    for (int s = 0; s < SPLITS; s++) {
        float4 v = *(const float4*)(partial + (size_t)(b * SPLITS + s) * DD + 4 * t);
        m.x += v.x; m.y += v.y; m.z += v.z; m.w += v.w;
    }
    const float invn = 1.0f / (float)NN;
    m.x *= invn; m.y *= invn; m.z *= invn; m.w *= invn;
    float ss = m.x * m.x + m.y * m.y + m.z * m.z + m.w * m.w;
    float tot = blockReduceSum256(ss, red);
    float inv = rsqrtf(fmaxf(tot, 1e-12f));
    float4 o = make_float4(m.x * inv, m.y * inv, m.z * inv, m.w * inv);
    *(float4*)(xnorm + b * DD + 4 * t) = o;
    *(float4*)(out + OFF_XN + b * DD + 4 * t) = o;
}

// ---------------- Kernel C: normalize prompt keys -----------------------------
__global__ __launch_bounds__(256) void norm_keys_kernel(const float* __restrict__ pk,
                                                        const float* __restrict__ pk2,
                                                        const int* __restrict__ task_id,
                                                        float* __restrict__ pkn,
                                                        float* __restrict__ pk2n,
                                                        float* __restrict__ out) {
    __shared__ float red[256];
    int j = blockIdx.x, t = threadIdx.x;
    int tid = task_id[0];
    const float* src;
    float* dst0;
    float* dst1 = nullptr;
    if (j < NCPT) {                    // rows sl..sl+9, sl = task_id*NCPT
        src = pk + (size_t)(tid * NCPT + j) * DD;
        dst0 = pkn + j * DD;
        dst1 = out + OFF_PKN + j * DD;
    } else {                           // prompt_key2[task_id]
        src = pk2 + (size_t)tid * DD;
        dst0 = pk2n;
    }
    float4 v = *(const float4*)(src + 4 * t);
    float ss = v.x * v.x + v.y * v.y + v.z * v.z + v.w * v.w;
    float tot = blockReduceSum256(ss, red);
    float inv = rsqrtf(fmaxf(tot, 1e-12f));
    float4 o = make_float4(v.x * inv, v.y * inv, v.z * inv, v.w * inv);
    *(float4*)(dst0 + 4 * t) = o;
    if (dst1) *(float4*)(dst1 + 4 * t) = o;
}

// ---------------- Kernel D: per-b sims, top-3, bpr, scalar partials -----------
__global__ __launch_bounds__(256) void per_batch_kernel(const float* __restrict__ xnorm,
                                                        const float* __restrict__ pkn,
                                                        const float* __restrict__ pk2n,
                                                        const float* __restrict__ prompt_key,
                                                        const int* __restrict__ y,
                                                        const float* __restrict__ desc,
                                                        float* __restrict__ bpr,
                                                        float* __restrict__ rs2p,
                                                        float* __restrict__ rsp,
                                                        float* __restrict__ out) {
    __shared__ float red[256];
    __shared__ float simsh[NCPT];
    __shared__ float stkv[TOPK];
    __shared__ int   stki[TOPK];
    int b = blockIdx.x, t = threadIdx.x;
    float4 x4 = *(const float4*)(xnorm + b * DD + 4 * t);

    for (int j = 0; j < NCPT; j++) {
        const float* p = pkn + j * DD + 4 * t;
        float v = x4.x * p[0] + x4.y * p[1] + x4.z * p[2] + x4.w * p[3];
        float s = blockReduceSum256(v, red);
        if (t == 0) simsh[j] = s;
    }
    {
        const float* p = pk2n + 4 * t;
        float v = x4.x * p[0] + x4.y * p[1] + x4.z * p[2] + x4.w * p[3];
        float s = blockReduceSum256(v, red);
        if (t == 0) rs2p[b] = s;
    }
    {
        int yb = y[b];
        float4 p4 = *(const float4*)(prompt_key + (size_t)yb * DD + 4 * t);
        float dv = x4.x * p4.x + x4.y * p4.y + x4.z * p4.z + x4.w * p4.w;
        float sv = p4.x * p4.x + p4.y * p4.y + p4.z * p4.z + p4.w * p4.w;
        float dY = blockReduceSum256(dv, red);
        float sY = blockReduceSum256(sv, red);
        if (t == 0) rsp[b] = dY * rsqrtf(fmaxf(sY, 1e-12f));
    }
    __syncthreads();
    if (t < NCPT) out[OFF_SIM + b * NCPT + t] = simsh[t];
    if (t == 0) {   // top-3, descending, ties -> lowest index (strict >)
        bool taken[NCPT];
        for (int j = 0; j < NCPT; j++) taken[j] = false;
        for (int k = 0; k < TOPK; k++) {
            float best = -3.4e38f; int bi = 0;
            for (int j = 0; j < NCPT; j++)
                if (!taken[j] && simsh[j] > best) { best = simsh[j]; bi = j; }
            taken[bi] = true; stkv[k] = best; stki[k] = bi;
        }
    }
    __syncthreads();
    if (t < HD) {
        float acc = 0.f;
        for (int k = 0; k < TOPK; k++) acc += stkv[k] * desc[stki[k] * HD + t];
        bpr[b * HD + t] = acc;
        out[OFF_DESC + b * HD + t] = acc;
    }
}

// ---------------- Kernel E: deterministic scalar finalize ---------------------
__global__ void scalar_kernel(const float* __restrict__ rs2p,
                              const float* __restrict__ rsp,
                              float* __restrict__ out) {
    if (threadIdx.x == 0 && blockIdx.x == 0) {
        float a = 0.f, c = 0.f;
        for (int i = 0; i < BB; i++) { a += rs2p[i]; c += rsp[i]; }
        out[OFF_RS2] = a / (float)BB;
        out[OFF_RS]  = c / (float)BB;
    }
}

// ---------------- Kernel F: fp32 WMMA attention chains ------------------------
// One wave per (kv, h, g): (32x64 bpr) @ v_w[h,g] (64x64) @ proj_w[h,g] (64x64) + bias
__device__ __forceinline__ v8f wmma_f32(v2f a, v2f b, v8f c) {
    return __builtin_amdgcn_wmma_f32_16x16x4_f32(false, a, false, b, (short)0, c,
                                                 false, false);
}

__global__ __launch_bounds__(32) void attn_wmma_kernel(const float* __restrict__ bpr,
                                                       const float* __restrict__ k_qkv,
                                                       const float* __restrict__ k_proj,
                                                       const float* __restrict__ k_bias,
                                                       const float* __restrict__ v_qkv,
                                                       const float* __restrict__ v_proj,
                                                       const float* __restrict__ v_bias,
                                                       float* __restrict__ out) {
    __shared__ float vsh[BB * HD];     // 32x64 intermediate, 8 KB

    int blk = blockIdx.x;              // kv*48 + h*3 + g
    int kv = blk / (HEADS * GRP);
    int hg = blk % (HEADS * GRP);
    int h = hg / GRP, g = hg % GRP;

    const float* qkv = kv ? v_qkv : k_qkv;
    const float* prj = kv ? v_proj : k_proj;
    const float* bia = kv ? v_bias : k_bias;
    const float* w1 = qkv + (size_t)hg * HD * (3 * HD) + 2 * HD;  // [k*192 + n], e-slice 128..191
    const float* w2 = prj + (size_t)hg * HD * HD;                  // [e*64 + f]
    const float* bb = bia + hg * HD;

    int lane = threadIdx.x;
    int m_lo = lane & 15;              // row (A/M), col (B/N) within 16
    int kh = (lane >> 4) * 2;          // K sub-offset for A/B fragments
    int mhi = (lane >> 4) * 8;         // C/D row offset

    // ---- GEMM 1: v = bpr(32x64) @ w1(64x64) ----
    v8f acc[2][4] = {};
#pragma unroll
    for (int kk = 0; kk < 16; kk++) {
        int k0 = kk * 4 + kh;
        v2f a0 = *(const v2f*)(bpr + (m_lo) * HD + k0);
        v2f a1 = *(const v2f*)(bpr + (16 + m_lo) * HD + k0);
#pragma unroll
        for (int ni = 0; ni < 4; ni++) {
            v2f bf;
            bf.x = w1[(k0) * (3 * HD) + ni * 16 + m_lo];
            bf.y = w1[(k0 + 1) * (3 * HD) + ni * 16 + m_lo];
            acc[0][ni] = wmma_f32(a0, bf, acc[0][ni]);
            acc[1][ni] = wmma_f32(a1, bf, acc[1][ni]);
        }
    }

    // Restage C-layout -> row-major LDS (then read back in A-layout)
#pragma unroll
    for (int mi = 0; mi < 2; mi++)
#pragma unroll
        for (int ni = 0; ni < 4; ni++)
#pragma unroll
            for (int r = 0; r < 8; r++)
                vsh[(mi * 16 + r + mhi) * HD + ni * 16 + m_lo] = acc[mi][ni][r];
    __syncthreads();

    // ---- GEMM 2: out = v(32x64) @ w2(64x64) + bias ----
    v8f acc2[2][4] = {};
#pragma unroll
    for (int kk = 0; kk < 16; kk++) {
        int k0 = kk * 4 + kh;
        v2f a0 = *(const v2f*)(&vsh[(m_lo) * HD + k0]);
        v2f a1 = *(const v2f*)(&vsh[(16 + m_lo) * HD + k0]);
#pragma unroll
        for (int ni = 0; ni < 4; ni++) {
            v2f bf;
            bf.x = w2[(k0) * HD + ni * 16 + m_lo];
            bf.y = w2[(k0 + 1) * HD + ni * 16 + m_lo];
            acc2[0][ni] = wmma_f32(a0, bf, acc2[0][ni]);
            acc2[1][ni] = wmma_f32(a1, bf, acc2[1][ni]);
        }
    }

#pragma unroll
    for (int mi = 0; mi < 2; mi++)
#pragma unroll
        for (int ni = 0; ni < 4; ni++)
#pragma unroll
            for (int r = 0; r < 8; r++) {
                int brow = mi * 16 + r + mhi;
                int f = ni * 16 + m_lo;
                out[OFF_BP + ((((size_t)kv * BB + brow) * HEADS + h) * GRP + g) * HD + f] =
                    acc2[mi][ni][r] + bb[f];
            }
}

extern "C" void kernel_launch(void* const* d_in, const int* in_sizes, int n_in,
                              void* d_out, int out_size, void* d_ws, size_t ws_size,
                              hipStream_t stream) {
    const float* x_embed   = (const float*)d_in[0];
    const int*   y         = (const int*)d_in[1];
    const int*   task_id   = (const int*)d_in[2];
    const float* prompt_key  = (const float*)d_in[3];
    const float* prompt_key2 = (const float*)d_in[4];
    const float* desc_embed  = (const float*)d_in[5];
    const float* k_qkv  = (const float*)d_in[6];
    const float* k_proj = (const float*)d_in[7];
    const float* k_bias = (const float*)d_in[8];
    const float* v_qkv  = (const float*)d_in[9];
    const float* v_proj = (const float*)d_in[10];
    const float* v_bias = (const float*)d_in[11];
    float* out = (float*)d_out;
    float* ws  = (float*)d_ws;

    mean_partial_kernel<<<BB * SPLITS, 256, 0, stream>>>(x_embed, ws + WS_PART);
    finalize_norm_kernel<<<BB, 256, 0, stream>>>(ws + WS_PART, ws + WS_XN, out);
    norm_keys_kernel<<<NCPT + 1, 256, 0, stream>>>(prompt_key, prompt_key2, task_id,
                                                   ws + WS_PKN, ws + WS_PK2, out);
    per_batch_kernel<<<BB, 256, 0, stream>>>(ws + WS_XN, ws + WS_PKN, ws + WS_PK2,
                                             prompt_key, y, desc_embed,
                                             ws + WS_BPR, ws + WS_RS2P, ws + WS_RSP, out);
    scalar_kernel<<<1, 32, 0, stream>>>(ws + WS_RS2P, ws + WS_RSP, out);
    attn_wmma_kernel<<<2 * HEADS * GRP, 32, 0, stream>>>(ws + WS_BPR, k_qkv, k_proj, k_bias,
                                                         v_qkv, v_proj, v_bias, out);
}